// BerryPhaseMultimodalComparator_89644557402423
// MI455X (gfx1250) — compile-verified
//
#include <hip/hip_runtime.h>
#include <math.h>

// ---------------------------------------------------------------------------
// BerryPhaseMultimodalComparator for MI455X (gfx1250, wave32)
//
// Fully fused, latency-oriented: latents never touch HBM. The K<=4 latent
// GEMM runs on V_WMMA_F32_16X16X4_F32 with transposed roles
// (A = W^T tile [16 cols x 4K], B = x^T [4K x 16 rows]) so the C layout puts
// two complete quaternions of one row into each lane's accumulators.
// Ordered quaternion chain product = shuffle/LDS tree reduction, with
// branch-free operand selection (cndmask) instead of divergent qmul paths.
// ---------------------------------------------------------------------------

typedef float v2f __attribute__((ext_vector_type(2)));
typedef float v8f __attribute__((ext_vector_type(8)));

struct Quat { float w, x, y, z; };

__device__ __forceinline__ Quat qmul(const Quat a, const Quat b) {
  Quat r;
  r.w = a.w*b.w - a.x*b.x - a.y*b.y - a.z*b.z;
  r.x = a.w*b.x + a.x*b.w + a.y*b.z - a.z*b.y;
  r.y = a.w*b.y - a.x*b.z + a.y*b.w + a.z*b.x;
  r.z = a.w*b.z + a.x*b.y - a.y*b.x + a.z*b.w;
  return r;
}

__device__ __forceinline__ float qnorm2(const Quat q) {
  return q.w*q.w + q.x*q.x + q.y*q.y + q.z*q.z;
}

// q / (|q| + eps) with raw hardware sqrt/rcp (v_sqrt_f32 / v_rcp_f32)
__device__ __forceinline__ Quat qnorm(Quat q) {
  float n   = __builtin_amdgcn_sqrtf(qnorm2(q));
  float inv = __builtin_amdgcn_rcpf(n + 1e-8f);
  q.w *= inv; q.x *= inv; q.y *= inv; q.z *= inv;
  return q;
}

__device__ __forceinline__ Quat qshfl_xor(Quat q, int mask) {
  Quat r;
  r.w = __shfl_xor(q.w, mask, 32);
  r.x = __shfl_xor(q.x, mask, 32);
  r.y = __shfl_xor(q.y, mask, 32);
  r.z = __shfl_xor(q.z, mask, 32);
  return r;
}

// branch-free per-component select (lowers to v_cndmask, no EXEC branching)
__device__ __forceinline__ Quat qsel(bool c, const Quat a, const Quat b) {
  Quat r;
  r.w = c ? a.w : b.w;
  r.x = c ? a.x : b.x;
  r.y = c ? a.y : b.y;
  r.z = c ? a.z : b.z;
  return r;
}

#define WG_PER_SEG 32   // ordered chunks per (modality,batch) segment
#define NSEG 24         // 3 modalities * 8 batches
#define D 256           // MANIFOLD_DIM

// ---------------------------------------------------------------------------
// Kernel 1: per-chunk ordered quaternion partial products (WMMA GEMM fused)
// grid = NSEG*WG_PER_SEG blocks of 256 threads (8 waves).
// ---------------------------------------------------------------------------
__global__ void __launch_bounds__(256)
berry_partials_kernel(const float* __restrict__ vis, const float* __restrict__ txt,
                      const float* __restrict__ gen,
                      const float* __restrict__ Wv, const float* __restrict__ bv,
                      const float* __restrict__ Wt, const float* __restrict__ bt,
                      const float* __restrict__ Wg, const float* __restrict__ bg,
                      float* __restrict__ outq)
{
  const int wg    = blockIdx.x;
  const int seg   = wg / WG_PER_SEG;
  const int chunk = wg % WG_PER_SEG;
  const int m = seg >> 3;       // 0=vision 1=text 2=genomic
  const int b = seg & 7;

  const float* x    = (m == 0) ? vis : ((m == 1) ? txt : gen);
  const float* W    = (m == 0) ? Wv  : ((m == 1) ? Wt  : Wg);
  const float* bias = (m == 0) ? bv  : ((m == 1) ? bt  : bg);
  const int K = (m == 0) ? 3 : ((m == 1) ? 1 : 4);
  const int R = (m == 1) ? 8192 : 4096;     // rows per batch
  const int rowsPerWg = R / WG_PER_SEG;     // 128 or 256
  const int nb = rowsPerWg >> 7;            // 128-row blocks per workgroup

  __shared__ float Wp[4][D];   // K-padded weights (A operand source)
  __shared__ float bsh[D];     // bias (C init)
  __shared__ Quat  wq[8];      // per-wave partials

  const int tid = threadIdx.x;
  #pragma unroll
  for (int k = 0; k < 4; ++k)
    Wp[k][tid] = (k < K) ? W[k * D + tid] : 0.0f;
  bsh[tid] = bias[tid];
  __syncthreads();

  const int wave  = tid >> 5;
  const int lane  = tid & 31;
  const int lrow  = lane & 15;            // row-within-16 this lane serves
  const bool lowHalf = (lane < 16);
  const int k0    = lowHalf ? 0 : 2;      // K pair this half supplies
  const int m_off = lowHalf ? 0 : 8;      // C-matrix M offset for this half
  const size_t xbase = (size_t)b * (size_t)R;

  Quat wgProd = {1.0f, 0.0f, 0.0f, 0.0f};

  for (int blk = 0; blk < nb; ++blk) {
    const int r0  = chunk * rowsPerWg + blk * 128 + wave * 16;
    const int row = r0 + lrow;
    const float* xr = x + (xbase + (size_t)row) * (size_t)K;

    // B operand: B[k, n] = x[b, r0+n, k]; K0/K2 split across wave halves
    v2f bop;
    bop[0] = (k0     < K) ? xr[k0]     : 0.0f;
    bop[1] = (k0 + 1 < K) ? xr[k0 + 1] : 0.0f;

    Quat P = {1.0f, 0.0f, 0.0f, 0.0f};  // ordered product over j for this row

    #pragma unroll 2
    for (int cb = 0; cb < D; cb += 16) {
      // A operand: A[mcol, k] = W[k, cb+mcol]  (16 latent cols x 4 K)
      v2f aop;
      aop[0] = Wp[k0][cb + lrow];
      aop[1] = Wp[k0 + 1][cb + lrow];

      // C init = bias (lat = W^T x + b)
      v8f c;
      #pragma unroll
      for (int r = 0; r < 8; ++r) c[r] = bsh[cb + m_off + r];

      c = __builtin_amdgcn_wmma_f32_16x16x4_f32(
              false, aop, false, bop, (short)0, c, false, false);

      // lane<16 holds quats j0,j0+1 of its row; lane>=16 holds j0+2,j0+3
      Quat qa; qa.w = c[0]; qa.x = c[1]; qa.y = c[2]; qa.z = c[3];
      Quat qb; qb.w = c[4]; qb.x = c[5]; qb.y = c[6]; qb.z = c[7];

      // fused pair-normalization: (qa/(|qa|+e)) * (qb/(|qb|+e))
      //                         = (qa*qb) * rcp((|qa|+e)*(|qb|+e))
      float na  = __builtin_amdgcn_sqrtf(qnorm2(qa));
      float nb2 = __builtin_amdgcn_sqrtf(qnorm2(qb));
      float inv = __builtin_amdgcn_rcpf((na + 1e-8f) * (nb2 + 1e-8f));
      Quat t = qmul(qa, qb);
      t.w *= inv; t.x *= inv; t.y *= inv; t.z *= inv;

      // merge the two halves (ordered): (j0 j0+1) * (j0+2 j0+3)
      // branch-free: pick operand order with cndmask, single qmul
      Quat o  = qshfl_xor(t, 16);
      Quat lo = qsel(lowHalf, t, o);
      Quat hi = qsel(lowHalf, o, t);
      t = qmul(lo, hi);                 // both halves end identical

      P = qmul(P, t);                   // ascending j within row
    }

    // ordered fold of 16 consecutive rows across lanes (halves duplicated)
    #pragma unroll
    for (int s = 1; s < 16; s <<= 1) {
      Quat o = qshfl_xor(P, s);
      const bool upper = (lane & s) != 0;
      Quat lo = qsel(upper, o, P);
      Quat hi = qsel(upper, P, o);
      P = qmul(lo, hi);
    }

    if (lane == 0) wq[wave] = P;
    __syncthreads();
    if (tid == 0) {
      #pragma unroll
      for (int w = 0; w < 8; ++w) wgProd = qmul(wgProd, wq[w]);
    }
    __syncthreads();
  }

  if (tid == 0) {
    float* dst = outq + (size_t)wg * 4;
    dst[0] = wgProd.w; dst[1] = wgProd.x; dst[2] = wgProd.y; dst[3] = wgProd.z;
  }
}

// ---------------------------------------------------------------------------
// Kernel 2: holomorphic audit partials. g[b,row,j] = (batch-diff of x) dot
// wsum[:,j] (bias cancels). 192 workgroups: 0..63 vision, 64..191 text.
// ---------------------------------------------------------------------------
__global__ void __launch_bounds__(256)
audit_partials_kernel(const float* __restrict__ vis, const float* __restrict__ txt,
                      const float* __restrict__ Wv, const float* __restrict__ Wt,
                      float* __restrict__ part)
{
  const int wg = blockIdx.x;
  const int m  = (wg < 64) ? 0 : 1;
  const float* x = (m == 0) ? vis : txt;
  const float* W = (m == 0) ? Wv  : Wt;
  const int K   = (m == 0) ? 3 : 1;
  const int R   = (m == 0) ? 4096 : 8192;
  const int nwg = (m == 0) ? 64 : 128;
  const int lwg = (m == 0) ? wg : wg - 64;

  __shared__ float wsum[4][64];
  __shared__ float red[256];
  const int tid = threadIdx.x;
  if (tid < 64) {
    #pragma unroll
    for (int k = 0; k < 4; ++k) {
      float s = 0.0f;
      if (k < K) {
        #pragma unroll
        for (int c = 0; c < 4; ++c) s += W[k * D + tid * 4 + c];
      }
      wsum[k][tid] = s;
    }
  }
  __syncthreads();

  const long bstride = (long)R * (long)K;   // floats between batches
  const int items = 8 * R;
  float acc = 0.0f;

  for (int i = lwg * 256 + tid; i < items; i += nwg * 256) {
    const int b = i / R;
    const int row = i - b * R;
    const float* xp = x + (long)b * bstride + (long)row * K;
    float dx[4] = {0.f, 0.f, 0.f, 0.f};
    for (int k = 0; k < K; ++k) {
      if (b == 0)        dx[k] = xp[bstride + k] - xp[k];
      else if (b == 7)   dx[k] = xp[k] - xp[k - bstride];
      else               dx[k] = 0.5f * (xp[bstride + k] - xp[k - bstride]);
    }
    #pragma unroll 8
    for (int j = 0; j < 64; ++j) {
      float g = dx[0] * wsum[0][j] + dx[1] * wsum[1][j] +
                dx[2] * wsum[2][j] + dx[3] * wsum[3][j];
      acc += fabsf(g);
    }
  }

  red[tid] = acc;
  __syncthreads();
  for (int s = 128; s > 0; s >>= 1) {
    if (tid < s) red[tid] += red[tid + s];
    __syncthreads();
  }
  if (tid == 0) part[wg] = red[0];
}

// ---------------------------------------------------------------------------
// Kernel 3: finalize. 1 block, 32 threads. Ordered combine of chunk partials,
// append q0, phi = 2*acos(clip(w)), then interference / scores / residuals.
// ---------------------------------------------------------------------------
__global__ void __launch_bounds__(32)
finalize_kernel(const float* __restrict__ vis, const float* __restrict__ txt,
                const float* __restrict__ gen,
                const float* __restrict__ Wv, const float* __restrict__ bv,
                const float* __restrict__ Wt, const float* __restrict__ bt,
                const float* __restrict__ Wg, const float* __restrict__ bg,
                const float* __restrict__ ws, float* __restrict__ out)
{
  __shared__ float phis[NSEG];
  const int tid = threadIdx.x;

  if (tid < NSEG) {
    const int seg = tid;
    const int m = seg >> 3, b = seg & 7;
    const float* x    = (m == 0) ? vis : ((m == 1) ? txt : gen);
    const float* W    = (m == 0) ? Wv  : ((m == 1) ? Wt  : Wg);
    const float* bias = (m == 0) ? bv  : ((m == 1) ? bt  : bg);
    const int K = (m == 0) ? 3 : ((m == 1) ? 1 : 4);
    const int R = (m == 1) ? 8192 : 4096;

    Quat p = {1.0f, 0.0f, 0.0f, 0.0f};
    for (int c = 0; c < WG_PER_SEG; ++c) {
      const float* q = ws + (size_t)(seg * WG_PER_SEG + c) * 4;
      Quat t; t.w = q[0]; t.x = q[1]; t.y = q[2]; t.z = q[3];
      p = qmul(p, t);
    }

    // q0 = normalized first quaternion of the segment (row 0, cols 0..3)
    const float* x0 = x + (size_t)b * (size_t)R * (size_t)K;
    float q0c[4];
    #pragma unroll
    for (int c = 0; c < 4; ++c) {
      float s = bias[c];
      for (int k = 0; k < K; ++k) s += x0[k] * W[k * D + c];
      q0c[c] = s;
    }
    Quat q0; q0.w = q0c[0]; q0.x = q0c[1]; q0.y = q0c[2]; q0.z = q0c[3];
    p = qmul(p, qnorm(q0));

    float wclip = fminf(fmaxf(p.w, -1.0f), 1.0f);
    phis[seg] = 2.0f * acosf(wclip);
  }
  __syncthreads();

  if (tid == 0) {
    float mv = 0.0f, mt = 0.0f;
    for (int i = 0; i < 8; ++i) { mv += phis[i]; mt += phis[8 + i]; }
    mv *= 0.125f; mt *= 0.125f;
    const float inter = fabsf(mv - mt);
    for (int i = 0; i < 8; ++i)
      out[i] = expf(-fabsf(phis[16 + i] - inter));
    out[8] = inter;

    const float* ap = ws + NSEG * WG_PER_SEG * 4;  // audit partials
    float sv = 0.0f, st = 0.0f;
    for (int i = 0; i < 64; ++i)  sv += ap[i];
    for (int i = 0; i < 128; ++i) st += ap[64 + i];
    out[9]  = sv / (8.0f * 4096.0f * 64.0f);
    out[10] = st / (8.0f * 8192.0f * 64.0f);
  }
}

// ---------------------------------------------------------------------------
extern "C" void kernel_launch(void* const* d_in, const int* in_sizes, int n_in,
                              void* d_out, int out_size, void* d_ws, size_t ws_size,
                              hipStream_t stream) {
  const float* vis = (const float*)d_in[0];
  const float* txt = (const float*)d_in[1];
  const float* gen = (const float*)d_in[2];
  const float* Wv  = (const float*)d_in[3];
  const float* bv  = (const float*)d_in[4];
  const float* Wt  = (const float*)d_in[5];
  const float* bt  = (const float*)d_in[6];
  const float* Wg  = (const float*)d_in[7];
  const float* bg  = (const float*)d_in[8];
  float* out = (float*)d_out;
  float* wsf = (float*)d_ws;                    // [0,3072): quat partials
  float* auditp = wsf + NSEG * WG_PER_SEG * 4;  // [3072,3264): audit partials

  berry_partials_kernel<<<NSEG * WG_PER_SEG, 256, 0, stream>>>(
      vis, txt, gen, Wv, bv, Wt, bt, Wg, bg, wsf);
  audit_partials_kernel<<<192, 256, 0, stream>>>(vis, txt, Wv, Wt, auditp);
  finalize_kernel<<<1, 32, 0, stream>>>(
      vis, txt, gen, Wv, bv, Wt, bt, Wg, bg, wsf, out);
}